// ScientificImagingObservationModel_71322226917785
// MI455X (gfx1250) — compile-verified
//
#include <hip/hip_runtime.h>
#include <math.h>

// Problem constants from the reference harness: B=8, C=2, D=256.
#define DVOL 256

__global__ __launch_bounds__(256) void slice_extract_phase_kernel(
    const float* __restrict__ vol,    // (B, 2, D, D, D)
    const float* __restrict__ rot,    // (B, 3, 3)
    const float* __restrict__ trans,  // (B, 2)
    float* __restrict__ out)          // (B, 2, D, D)
{
    const int D = DVOL;
    // Output pixel (i, j): i = row (dim 2), j = col (dim 3, fastest -> coalesced stores).
    const int j = blockIdx.x * 16 + threadIdx.x;
    const int i = blockIdx.y * 16 + threadIdx.y;
    const int b = blockIdx.z;

    // Rotation matrix: uniform per block -> compiler scalarizes these loads.
    const float* R = rot + b * 9;
    const float r00 = R[0], r01 = R[1];
    const float r10 = R[3], r11 = R[4];
    const float r20 = R[6], r21 = R[7];

    // Output has the H/W transpose folded in: out[b,ch,i,j] = proj[b,ch,h=j,w=i].
    // Grid coords: gx = coords1d[w] = i-128, gy = coords1d[h] = j-128.
    const float gx = (float)i - 128.0f;
    const float gy = (float)j - 128.0f;

    // sl coordinate d = gx*R[d,0] + gy*R[d,1] + D/2   (coord 0 = x = last vol axis)
    float xc = fmaf(gx, r00, fmaf(gy, r01, 128.0f));
    float yc = fmaf(gx, r10, fmaf(gy, r11, 128.0f));
    float zc = fmaf(gx, r20, fmaf(gy, r21, 128.0f));

    // Clamp to [0, D-1] as in the reference.
    xc = fminf(fmaxf(xc, 0.0f), 255.0f);
    yc = fminf(fmaxf(yc, 0.0f), 255.0f);
    zc = fminf(fmaxf(zc, 0.0f), 255.0f);

    // Boundary-folded cell: x0 = min(floor(x), D-2), fx = x - x0.
    // Identical interpolation result to (x0=floor, x1=min(x0+1,D-1)) incl. x==255.
    const int x0 = min((int)xc, 254);
    const int y0 = min((int)yc, 254);
    const int z0 = min((int)zc, 254);
    const float fx = xc - (float)x0;
    const float fy = yc - (float)y0;
    const float fz = zc - (float)z0;

    // Per-lane 32-bit offsets on a per-batch 64-bit scalar base (SADDR + voffset).
    const unsigned chS = (unsigned)D * D * D;               // 16.7M elems (64 MB)
    const float* __restrict__ v0 = vol + (size_t)b * 2u * chS;  // re channel
    const float* __restrict__ v1 = v0 + chS;                    // im channel
    const unsigned o00 = ((unsigned)z0 * D + (unsigned)y0) * D + (unsigned)x0; // (z0,y0,x0)
    const unsigned o01 = o00 + D;                            // (z0,y1)
    const unsigned o10 = o00 + (unsigned)D * D;              // (z1,y0)
    const unsigned o11 = o10 + D;                            // (z1,y1)

    // Issue all 16 independent gathers up front; dependent math comes after,
    // so the wave accumulates LOADcnt and overlaps all fetch latency.
    const float a000 = v0[o00], a001 = v0[o00 + 1];
    const float a010 = v0[o01], a011 = v0[o01 + 1];
    const float a100 = v0[o10], a101 = v0[o10 + 1];
    const float a110 = v0[o11], a111 = v0[o11 + 1];
    const float b000 = v1[o00], b001 = v1[o00 + 1];
    const float b010 = v1[o01], b011 = v1[o01 + 1];
    const float b100 = v1[o10], b101 = v1[o10 + 1];
    const float b110 = v1[o11], b111 = v1[o11 + 1];

    // Trilinear reduce (lerp along x, then y, then z).
    const float ax00 = fmaf(fx, a001 - a000, a000);
    const float ax01 = fmaf(fx, a011 - a010, a010);
    const float ax10 = fmaf(fx, a101 - a100, a100);
    const float ax11 = fmaf(fx, a111 - a110, a110);
    const float ay0  = fmaf(fy, ax01 - ax00, ax00);
    const float ay1  = fmaf(fy, ax11 - ax10, ax10);
    float re         = fmaf(fz, ay1 - ay0, ay0);

    const float bx00 = fmaf(fx, b001 - b000, b000);
    const float bx01 = fmaf(fx, b011 - b010, b010);
    const float bx10 = fmaf(fx, b101 - b100, b100);
    const float bx11 = fmaf(fx, b111 - b110, b110);
    const float by0  = fmaf(fy, bx01 - bx00, bx00);
    const float by1  = fmaf(fy, bx11 - bx10, bx10);
    float im         = fmaf(fz, by1 - by0, by0);

    // sqrt(D) = 16 exactly.
    re *= 16.0f;
    im *= 16.0f;

    // Fourier phase ramp: t0 = -trans[b,1]*D/2, t1 = -trans[b,0]*D/2,
    // ang = (-2*pi/D) * (t0*(i - D/2) + t1*(j - D/2)).
    const float t0 = -trans[b * 2 + 1] * 128.0f;
    const float t1 = -trans[b * 2 + 0] * 128.0f;
    const float ki = (float)(i - 128);
    const float kj = (float)(j - 128);
    const float ang = (-6.2831853071795864769f / 256.0f) * fmaf(t0, ki, t1 * kj);
    float sn, cs;
    sincosf(ang, &sn, &cs);  // accurate path; VALU is far from the bottleneck

    const size_t ob = (size_t)b * 2u * D * D + (size_t)i * D + (size_t)j;
    out[ob]                    = fmaf(re, cs, -im * sn);  // re' = re*cs - im*sn
    out[ob + (size_t)D * D]    = fmaf(re, sn,  im * cs);  // im' = re*sn + im*cs
}

extern "C" void kernel_launch(void* const* d_in, const int* in_sizes, int n_in,
                              void* d_out, int out_size, void* d_ws, size_t ws_size,
                              hipStream_t stream) {
    const float* vol   = (const float*)d_in[0];   // (B,2,D,D,D) fp32
    const float* rot   = (const float*)d_in[1];   // (B,3,3)     fp32
    const float* trans = (const float*)d_in[2];   // (B,2)       fp32
    float* out = (float*)d_out;                   // (B,2,D,D)   fp32

    const int B = in_sizes[1] / 9;                // = 8 for this harness
    dim3 block(16, 16, 1);                        // 256 threads = 8 wave32
    dim3 grid(DVOL / 16, DVOL / 16, B);           // 16 x 16 x 8 = 2048 blocks
    hipLaunchKernelGGL(slice_extract_phase_kernel, grid, block, 0, stream,
                       vol, rot, trans, out);
}